// CTCLoss_46875273068588
// MI455X (gfx1250) — compile-verified
//
#include <hip/hip_runtime.h>
#include <hip/hip_bf16.h>

typedef __attribute__((ext_vector_type(2))) float v2f;
typedef __attribute__((ext_vector_type(8))) float v8f;

#define CTC_B    128
#define CTC_T    128
#define CTC_C    6625
#define CTC_LMAX 25
#define CTC_S    51      // 2*Lmax+1
#define CTC_SP   52      // padded LDS stride
#define NEGF     (-1e30f)

__device__ __forceinline__ float lae(float a, float b) {
  // logaddexp with -1e30 as -inf surrogate (matches jnp.logaddexp to fp32)
  float mx = fmaxf(a, b);
  float mn = fminf(a, b);
  return mx + log1pf(__expf(mn - mx));
}

// ---------------------------------------------------------------------------
// Kernel 1: one block per (b,t) row of pred [16384 rows x 6625].
//   pass 1: row max (streams row HBM->L2)
//   pass 2: sum(exp(x-M)) with the adds accumulated on the matrix pipe via
//           v_wmma_f32_16x16x4_f32 (B = all-ones => exact f32 row sums);
//           row re-read hits L2/WGP$, so HBM traffic is 434 MB total.
//   tail:   gather the 51 extended-label log-probs (x[ext[s]] - lse) while
//           the row is cache-hot, so the DP kernel never touches pred.
// ---------------------------------------------------------------------------
__global__ __launch_bounds__(256) void k_lse_gather(
    const float* __restrict__ pred, const int* __restrict__ label,
    float* __restrict__ lpext_out) {
  const int row = blockIdx.x;            // row = b*T + t
  const int tid = threadIdx.x;
  const float* rp = pred + (size_t)row * CTC_C;

  // ---- pass 1: block max ----
  float m = NEGF;
#pragma unroll
  for (int it = 0; it < 28; ++it) {      // 28*256 = 7168 >= 6625
    int idx = tid + it * 256;
    if (idx < CTC_C) m = fmaxf(m, rp[idx]);
  }
#pragma unroll
  for (int off = 16; off > 0; off >>= 1) m = fmaxf(m, __shfl_xor(m, off, 32));

  __shared__ float smax[8];
  __shared__ float ssum[8];
  __shared__ float s_lse;
  const int wid = tid >> 5;
  if ((tid & 31) == 0) smax[wid] = m;
  __syncthreads();
  float M = smax[0];
#pragma unroll
  for (int w = 1; w < 8; ++w) M = fmaxf(M, smax[w]);

  // ---- pass 2: sum of exp(x-M), WMMA-accumulated; EXEC kept all-ones ----
  v8f acc = {};                          // C accumulator, starts at 0
  const v2f ones = {1.0f, 1.0f};
#pragma unroll
  for (int it = 0; it < 7; ++it) {       // uniform trip count for every lane
    int i0 = tid + (it * 4 + 0) * 256;
    int i1 = tid + (it * 4 + 1) * 256;
    int i2 = tid + (it * 4 + 2) * 256;
    int i3 = tid + (it * 4 + 3) * 256;
    float x0 = rp[i0 < CTC_C ? i0 : CTC_C - 1];
    float x1 = rp[i1 < CTC_C ? i1 : CTC_C - 1];
    float x2 = rp[i2 < CTC_C ? i2 : CTC_C - 1];
    float x3 = rp[i3 < CTC_C ? i3 : CTC_C - 1];
    float e0 = (i0 < CTC_C) ? __expf(x0 - M) : 0.0f;
    float e1 = (i1 < CTC_C) ? __expf(x1 - M) : 0.0f;
    float e2 = (i2 < CTC_C) ? __expf(x2 - M) : 0.0f;
    float e3 = (i3 < CTC_C) ? __expf(x3 - M) : 0.0f;
    v2f a0 = {e0, e1};
    v2f a1 = {e2, e3};
    // D = A x ones + C : exact f32 partial sums on the matrix pipe
    acc = __builtin_amdgcn_wmma_f32_16x16x4_f32(false, a0, false, ones,
                                                (short)0, acc, false, false);
    acc = __builtin_amdgcn_wmma_f32_16x16x4_f32(false, a1, false, ones,
                                                (short)0, acc, false, false);
  }
  // With B = ones, every D column holds all 16 row-sums, so
  // sum over all lanes/components == 16 * (wave total). Layout-agnostic.
  float t = acc[0] + acc[1] + acc[2] + acc[3] + acc[4] + acc[5] + acc[6] + acc[7];
#pragma unroll
  for (int off = 16; off > 0; off >>= 1) t += __shfl_xor(t, off, 32);
  t *= (1.0f / 16.0f);
  if ((tid & 31) == 0) ssum[wid] = t;
  __syncthreads();
  if (tid == 0) {
    float S = 0.0f;
#pragma unroll
    for (int w = 0; w < 8; ++w) S += ssum[w];
    s_lse = M + __logf(S);
  }
  __syncthreads();

  // ---- gather extended-label log-probs while row is L2-hot ----
  if (tid < CTC_S) {
    const int b = row >> 7;                                  // row / T
    const int e = (tid & 1) ? label[b * CTC_LMAX + (tid >> 1)] : 0; // BLANK=0
    lpext_out[(size_t)row * CTC_S + tid] = rp[e] - s_lse;
  }
}

// ---------------------------------------------------------------------------
// Kernel 2: CTC forward DP, one block per batch. Stage lp_ext[T][S] into LDS
// (coalesced), then a single wave runs the 128-step alpha recursion with
// double-buffered LDS (per-wave DS ordering => no workgroup barriers needed).
// ---------------------------------------------------------------------------
__global__ __launch_bounds__(128) void k_ctc_dp(
    const float* __restrict__ lpext, const int* __restrict__ label,
    const int* __restrict__ lab_len, float* __restrict__ partial) {
  __shared__ float lp[CTC_T * CTC_SP];   // 128*52*4 = 26.6 KB
  __shared__ float bufA[CTC_S];
  __shared__ float bufB[CTC_S];
  __shared__ unsigned char skipS[CTC_S];
  const int b = blockIdx.x;
  const int tid = threadIdx.x;

  if (tid < CTC_S) {
    bool sk = false;
    if ((tid & 1) && tid >= 3) {         // odd s>=3: ext[s]!=0 always (labels>=1)
      sk = label[b * CTC_LMAX + (tid >> 1)] !=
           label[b * CTC_LMAX + (tid >> 1) - 1];
    }
    skipS[tid] = sk ? 1 : 0;
  }

  const float* src = lpext + (size_t)b * CTC_T * CTC_S;
  for (int i = tid; i < CTC_T * CTC_S; i += 128) {
    int tt = i / CTC_S;
    int ss = i - tt * CTC_S;
    lp[tt * CTC_SP + ss] = src[i];
  }
  __syncthreads();

  if (tid < 32) {                        // single-wave DP: lockstep, no barriers
    float* cur = bufA;
    float* nxt = bufB;
    for (int s = tid; s < CTC_S; s += 32)
      cur[s] = (s < 2) ? lp[s] : NEGF;   // alpha0
    __builtin_amdgcn_wave_barrier();
    for (int t = 1; t < CTC_T; ++t) {
      for (int s = tid; s < CTC_S; s += 32) {
        float a0 = cur[s];
        float a1 = (s >= 1) ? cur[s - 1] : NEGF;
        float a2 = (s >= 2) ? cur[s - 2] : NEGF;
        float r = lae(a0, a1);
        if (skipS[s]) r = lae(r, a2);
        nxt[s] = r + lp[t * CTC_SP + s];
      }
      __builtin_amdgcn_wave_barrier();
      float* tm = cur; cur = nxt; nxt = tm;
    }
    if (tid == 0) {
      int L = lab_len[b];                // 1..25
      float loss = -lae(cur[2 * L], cur[2 * L - 1]);
      partial[b] = loss / ((float)L * (float)CTC_B);  // pre-scaled for mean
    }
  }
}

// ---------------------------------------------------------------------------
// Kernel 3: deterministic 128-way sum -> scalar output.
// ---------------------------------------------------------------------------
__global__ __launch_bounds__(128) void k_final(const float* __restrict__ partial,
                                               float* __restrict__ out) {
  const int tid = threadIdx.x;
  float v = partial[tid];
#pragma unroll
  for (int off = 16; off > 0; off >>= 1) v += __shfl_xor(v, off, 32);
  __shared__ float s[4];
  if ((tid & 31) == 0) s[tid >> 5] = v;
  __syncthreads();
  if (tid == 0) out[0] = s[0] + s[1] + s[2] + s[3];
}

extern "C" void kernel_launch(void* const* d_in, const int* in_sizes, int n_in,
                              void* d_out, int out_size, void* d_ws, size_t ws_size,
                              hipStream_t stream) {
  const float* pred   = (const float*)d_in[0];   // [B,T,C] f32
  const int* label    = (const int*)d_in[1];     // [B,Lmax] i32
  const int* lab_len  = (const int*)d_in[2];     // [B] i32
  float* lpext   = (float*)d_ws;                              // BT*S floats (~3.34 MB)
  float* partial = lpext + (size_t)CTC_B * CTC_T * CTC_S;     // 128 floats

  k_lse_gather<<<CTC_B * CTC_T, 256, 0, stream>>>(pred, label, lpext);
  k_ctc_dp<<<CTC_B, 128, 0, stream>>>(lpext, label, lab_len, partial);
  k_final<<<1, 128, 0, stream>>>(partial, (float*)d_out);
}